// TrajectoryScore_20023137534194
// MI455X (gfx1250) — compile-verified
//
#include <hip/hip_runtime.h>
#include <hip/hip_bf16.h>

typedef __attribute__((ext_vector_type(2)))  float v2f;
typedef __attribute__((ext_vector_type(4)))  float v4f;
typedef __attribute__((ext_vector_type(8)))  float v8f;

#define BATCH   64
#define NELEM   (256 * 1024)          // T*O elements per batch
#define CHUNKS  64                    // blocks per batch -> 64 partials/batch
#define THREADS 256
#define EPT     (NELEM / (CHUNKS * THREADS))   // 16 elements per thread
#define MU_FACTOR 1.1f

// ---------------- Stage 1: streaming exp-norm partial sums ----------------
// Each thread owns 16 contiguous 3-vectors = 48 floats = 12 b128 loads.
// Per-thread base byte offset = elem*12 bytes, elem multiple of 16 -> 16B aligned.
__global__ __launch_bounds__(THREADS)
void trajscore_partial_kernel(const float* __restrict__ z,
                              const float* __restrict__ R,
                              float* __restrict__ partials) {
    const int chunk = blockIdx.x;
    const int batch = blockIdx.y;
    const int tid   = threadIdx.x;

    const float r  = R[batch];
    const float Bc = -0.5f / (r * r);

    const long long elem0 = (long long)batch * NELEM
                          + (long long)chunk * (THREADS * EPT)
                          + (long long)tid * EPT;
    const v4f* p = (const v4f*)(z + elem0 * 3);

    float acc = 0.0f;
#pragma unroll
    for (int g = 0; g < EPT / 4; ++g) {
        // 3 x b128 = 4 elements; stream with non-temporal hint (data read once)
        v4f f0 = __builtin_nontemporal_load(&p[g * 3 + 0]);
        v4f f1 = __builtin_nontemporal_load(&p[g * 3 + 1]);
        v4f f2 = __builtin_nontemporal_load(&p[g * 3 + 2]);

        float n0 = sqrtf(f0.x * f0.x + f0.y * f0.y + f0.z * f0.z);
        float n1 = sqrtf(f0.w * f0.w + f1.x * f1.x + f1.y * f1.y);
        float n2 = sqrtf(f1.z * f1.z + f1.w * f1.w + f2.x * f2.x);
        float n3 = sqrtf(f2.y * f2.y + f2.z * f2.z + f2.w * f2.w);

        acc += __expf(Bc * n0);
        acc += __expf(Bc * n1);
        acc += __expf(Bc * n2);
        acc += __expf(Bc * n3);
    }

    __shared__ float sm[THREADS];
    sm[tid] = acc;
    __syncthreads();
#pragma unroll
    for (int s = THREADS / 2; s > 0; s >>= 1) {
        if (tid < s) sm[tid] += sm[tid + s];
        __syncthreads();
    }
    if (tid == 0) partials[batch * CHUNKS + chunk] = sm[0];
}

// ---------------- Stage 2: WMMA 64-way exact reduction + tail math ----------------
// One wave32 per batch. 64 partials live as A(16x4) f32, B = ones(4x16):
// D[m,n] = sum_k A[m,k] exactly (1.0 * x products, f32 accum). Summing the 8
// accumulator VGPRs per lane leaves half-wave totals; one shfl_xor(16) finishes.
__global__ __launch_bounds__(32)
void trajscore_final_kernel(const float* __restrict__ partials,
                            const float* __restrict__ R,
                            const int*   __restrict__ num_obs_p,
                            float* __restrict__ out) {
    const int batch = blockIdx.x;
    const int lane  = threadIdx.x;   // 0..31, full wave active (EXEC all-ones)

    v2f a;
    a.x = partials[batch * CHUNKS + 2 * lane];
    a.y = partials[batch * CHUNKS + 2 * lane + 1];

    float total;
#if __has_builtin(__builtin_amdgcn_wmma_f32_16x16x4_f32)
    v2f b;
    b.x = 1.0f;
    b.y = 1.0f;
    v8f c = {};
    // 8 args: (neg_a, A, neg_b, B, c_mod, C, reuse_a, reuse_b)
    c = __builtin_amdgcn_wmma_f32_16x16x4_f32(false, a, false, b,
                                              (short)0, c, false, false);
    total = c[0] + c[1] + c[2] + c[3] + c[4] + c[5] + c[6] + c[7];
#else
    // Fallback: plain shuffle tree (keeps compile green if builtin is absent)
    total = a.x + a.y;
    total += __shfl_xor(total, 1, 32);
    total += __shfl_xor(total, 2, 32);
    total += __shfl_xor(total, 4, 32);
    total += __shfl_xor(total, 8, 32);
#endif
    total += __shfl_xor(total, 16, 32);   // combine half-wave sums

    if (lane == 0) {
        const float score = total;
        const float r   = R[batch];
        const float A   = 1.0f / (r * r);
        const float m2A = -2.0f * A;
        const float mean  = expm1f(m2A) / m2A;
        const float var   = (A / tanhf(A) - 1.0f) * mean * mean;
        const float n     = (float)(*num_obs_p);
        const float mu    = n * mean;
        const float sigma = sqrtf(n * var);
        out[batch]         = score;                              // score
        out[BATCH + batch] = (score - MU_FACTOR * mu) / sigma;   // t_score
    }
}

extern "C" void kernel_launch(void* const* d_in, const int* in_sizes, int n_in,
                              void* d_out, int out_size, void* d_ws, size_t ws_size,
                              hipStream_t stream) {
    const float* z        = (const float*)d_in[0];  // (64,256,1024,3) f32
    const float* R        = (const float*)d_in[1];  // (64,) f32
    const int*   num_obs  = (const int*)d_in[2];    // scalar int
    float*       out      = (float*)d_out;          // 128 floats: score | t_score
    float*       partials = (float*)d_ws;           // 64*64 floats = 16 KB

    dim3 grid1(CHUNKS, BATCH);
    trajscore_partial_kernel<<<grid1, THREADS, 0, stream>>>(z, R, partials);
    trajscore_final_kernel<<<BATCH, 32, 0, stream>>>(partials, R, num_obs, out);
}